// Detect_18786186952922
// MI455X (gfx1250) — compile-verified
//
#include <hip/hip_runtime.h>
#include <cstdint>
#include <cstddef>

#define BATCH   128
#define NPRIOR  8732
#define NCLS    21
#define TOPK    200
#define CONF_TH 0.01f
#define NMS_TH  0.45f
#define VAR0    0.1f
#define VAR1    0.2f
#define NT1     256
#define NT2     1024
#define NJ      9          // ceil(NPRIOR / NT2)

// ---------------------------------------------------------------------------
// CDNA5 async global->LDS copy (tracked by ASYNCcnt). VDST vgpr carries the
// LDS byte offset, VADDR pair carries the 64-bit global address (GV mode).
// ---------------------------------------------------------------------------
__device__ __forceinline__ uint32_t lds_off32(const void* p) {
  // Flat shared-aperture addresses keep the LDS byte offset in the low 32 bits.
  return (uint32_t)(uintptr_t)p;
}

__device__ __forceinline__ void async_copy_b128(uint32_t lds_byte_off, const void* gptr) {
  asm volatile("global_load_async_to_lds_b128 %0, %1, off"
               :
               : "v"(lds_byte_off), "v"((uint64_t)(uintptr_t)gptr)
               : "memory");
}

__device__ __forceinline__ void wait_async_zero() {
  asm volatile("s_wait_asynccnt 0x0" ::: "memory");
}

// ---------------------------------------------------------------------------
// Kernel 1: softmax score + argmax class + box decode, SoA output planes.
// ---------------------------------------------------------------------------
__global__ void __launch_bounds__(NT1) decode_score_kernel(
    const float* __restrict__ loc, const float* __restrict__ conf,
    const float* __restrict__ prior,
    float* __restrict__ x1p, float* __restrict__ y1p,
    float* __restrict__ x2p, float* __restrict__ y2p,
    float* __restrict__ scp, int* __restrict__ clp)
{
  const int idx = blockIdx.x * NT1 + threadIdx.x;
  if (idx >= BATCH * NPRIOR) return;
  const int p = idx % NPRIOR;

  // Softmax over 21 logits; only need max over classes 1..20 of e[c]/sum.
  const float* c = conf + (size_t)idx * NCLS;
  float v[NCLS];
  float m = -3.402823466e38f;
#pragma unroll
  for (int i = 0; i < NCLS; ++i) { v[i] = c[i]; m = fmaxf(m, v[i]); }
  float sum = 0.0f;
#pragma unroll
  for (int i = 0; i < NCLS; ++i) { v[i] = expf(v[i] - m); sum += v[i]; }
  float emax = -1.0f; int cbest = 0;
#pragma unroll
  for (int i = 1; i < NCLS; ++i) {
    if (v[i] > emax) { emax = v[i]; cbest = i - 1; }   // strict > -> first max
  }
  const float score = emax / sum;
  scp[idx] = (score > CONF_TH) ? score : -1.0f;
  clp[idx] = cbest;

  // Decode: xy = p_cxy + l_xy*0.1*p_wh ; wh = p_wh*exp(l_wh*0.2)
  const float4 l  = *(const float4*)(loc + (size_t)idx * 4);
  const float4 pr = *(const float4*)(prior + (size_t)p * 4);
  const float cx = pr.x + l.x * VAR0 * pr.z;
  const float cy = pr.y + l.y * VAR0 * pr.w;
  const float w  = pr.z * expf(l.z * VAR1);
  const float h  = pr.w * expf(l.w * VAR1);
  x1p[idx] = cx - 0.5f * w;
  y1p[idx] = cy - 0.5f * h;
  x2p[idx] = cx + 0.5f * w;
  y2p[idx] = cy + 0.5f * h;
}

// ---------------------------------------------------------------------------
// Kernel 2: one workgroup per batch image. Read-only box/class planes live in
// LDS (5 * 8732 * 4B = 174,640 B), staged via async B128 copies. Scores and
// each thread's own boxes are register-resident, so every NMS round is pure
// VALU + one 5-word LDS broadcast + a 2-barrier cross-wave argmax.
// ---------------------------------------------------------------------------
__global__ void __launch_bounds__(NT2) nms_kernel(
    const float* __restrict__ x1p, const float* __restrict__ y1p,
    const float* __restrict__ x2p, const float* __restrict__ y2p,
    const float* __restrict__ scp, const int* __restrict__ clp,
    float* __restrict__ out)
{
  extern __shared__ float smem[];
  float* sx1 = smem;
  float* sy1 = sx1 + NPRIOR;
  float* sx2 = sy1 + NPRIOR;
  float* sy2 = sx2 + NPRIOR;
  int*   scl = (int*)(sy2 + NPRIOR);
  __shared__ float rs[64];            // double-buffered (32 per parity)
  __shared__ int   ri[64];

  const int b   = blockIdx.x;
  const int tid = threadIdx.x;
  const size_t base = (size_t)b * NPRIOR;

  // ---- Stage the five read-only planes global -> LDS (async B128) --------
  const int nchunk = NPRIOR / 4;            // 8732 % 4 == 0 -> 2183 chunks
  const uint32_t ox1 = lds_off32(sx1), oy1 = lds_off32(sy1);
  const uint32_t ox2 = lds_off32(sx2), oy2 = lds_off32(sy2);
  const uint32_t ocl = lds_off32(scl);
  for (int cI = tid; cI < nchunk; cI += NT2) {
    const uint32_t bo = (uint32_t)cI * 16u;
    const size_t  go = base + (size_t)cI * 4;
    async_copy_b128(ox1 + bo, x1p + go);
    async_copy_b128(oy1 + bo, y1p + go);
    async_copy_b128(ox2 + bo, x2p + go);
    async_copy_b128(oy2 + bo, y2p + go);
    async_copy_b128(ocl + bo, clp + go);
  }

  // ---- Register-resident scores (global, coalesced; L2-resident) ---------
  float my_s[NJ];
#pragma unroll
  for (int j = 0; j < NJ; ++j) {
    const int i = tid + j * NT2;
    my_s[j] = (i < NPRIOR) ? scp[base + i] : -3.402823466e38f;
  }

  wait_async_zero();
  __syncthreads();

  // ---- Own boxes/classes into registers from LDS (stride-1024, no bank
  //      conflicts within a wave) -----------------------------------------
  float mx1[NJ], my1[NJ], mx2[NJ], my2[NJ];
  int   myc[NJ];
#pragma unroll
  for (int j = 0; j < NJ; ++j) {
    const int i = tid + j * NT2;
    if (i < NPRIOR) {
      mx1[j] = sx1[i]; my1[j] = sy1[i];
      mx2[j] = sx2[i]; my2[j] = sy2[i];
      myc[j] = scl[i];
    } else {
      mx1[j] = 0.0f; my1[j] = 0.0f; mx2[j] = 0.0f; my2[j] = 0.0f;
      myc[j] = -1;                  // never matches a real class (0..19)
    }
  }

  float* orow = out + (size_t)b * TOPK * 6;

  for (int it = 0; it < TOPK; ++it) {
    const int off = (it & 1) << 5;  // parity-selected reduction buffer

    // ---- local argmax over registers (ascending i -> first-max kept) -----
    float bs = -3.402823466e38f;
    int   bi = 0x7fffffff;
#pragma unroll
    for (int j = 0; j < NJ; ++j) {
      if (my_s[j] > bs) { bs = my_s[j]; bi = tid + j * NT2; }
    }
    // ---- wave32 reduce, tie-break on lowest index (jnp.argmax) -----------
#pragma unroll
    for (int d = 16; d > 0; d >>= 1) {
      const float s2 = __shfl_xor(bs, d, 32);
      const int   i2 = __shfl_xor(bi, d, 32);
      if (s2 > bs || (s2 == bs && i2 < bi)) { bs = s2; bi = i2; }
    }
    if ((tid & 31) == 0) { rs[off + (tid >> 5)] = bs; ri[off + (tid >> 5)] = bi; }
    __syncthreads();
    if (tid < 32) {
      bs = rs[off + tid]; bi = ri[off + tid];
#pragma unroll
      for (int d = 16; d > 0; d >>= 1) {
        const float s2 = __shfl_xor(bs, d, 32);
        const int   i2 = __shfl_xor(bi, d, 32);
        if (s2 > bs || (s2 == bs && i2 < bi)) { bs = s2; bi = i2; }
      }
      if (tid == 0) { rs[off] = bs; ri[off] = bi; }
    }
    __syncthreads();

    const float smax = rs[off];
    const int   imax = ri[off];
    // same-address LDS reads broadcast conflict-free
    const float bx1 = sx1[imax], by1 = sy1[imax];
    const float bx2 = sx2[imax], by2 = sy2[imax];
    const int   cmax = scl[imax];

    if (tid == 0) {
      if (smax > 0.0f) {
        orow[0] = bx1; orow[1] = by1; orow[2] = bx2; orow[3] = by2;
        orow[4] = smax; orow[5] = (float)cmax;
      } else {
#pragma unroll
        for (int k = 0; k < 6; ++k) orow[k] = 0.0f;
      }
    }
    orow += 6;

    // ---- suppression, pure registers (exactly mirrors the reference) -----
    const float area_i = (bx2 - bx1) * (by2 - by1);
#pragma unroll
    for (int j = 0; j < NJ; ++j) {
      const float xx1 = fmaxf(bx1, mx1[j]);
      const float yy1 = fmaxf(by1, my1[j]);
      const float xx2 = fminf(bx2, mx2[j]);
      const float yy2 = fminf(by2, my2[j]);
      const float inter = fmaxf(xx2 - xx1, 0.0f) * fmaxf(yy2 - yy1, 0.0f);
      const float areas = (mx2[j] - mx1[j]) * (my2[j] - my1[j]);
      const float iou = inter / (area_i + areas - inter);
      if (iou > NMS_TH && myc[j] == cmax) my_s[j] = -1.0f;
    }
    // No trailing barrier: next round writes the OTHER rs/ri buffer, and the
    // two in-round barriers bound skew to < 1 round.
  }
}

// ---------------------------------------------------------------------------
extern "C" void kernel_launch(void* const* d_in, const int* in_sizes, int n_in,
                              void* d_out, int out_size, void* d_ws, size_t ws_size,
                              hipStream_t stream) {
  (void)in_sizes; (void)n_in; (void)out_size; (void)ws_size;
  const float* loc   = (const float*)d_in[0];   // (B,P,4)
  const float* conf  = (const float*)d_in[1];   // (B*P,21)
  const float* prior = (const float*)d_in[2];   // (P,4)
  float* out = (float*)d_out;                   // (B,TOPK,6)

  const size_t NP = (size_t)BATCH * NPRIOR;
  float* x1p = (float*)d_ws;
  float* y1p = x1p + NP;
  float* x2p = y1p + NP;
  float* y2p = x2p + NP;
  float* scp = y2p + NP;
  int*   clp = (int*)(scp + NP);

  const int total = BATCH * NPRIOR;
  decode_score_kernel<<<(total + NT1 - 1) / NT1, NT1, 0, stream>>>(
      loc, conf, prior, x1p, y1p, x2p, y2p, scp, clp);

  const size_t shmem = (size_t)NPRIOR * 5 * sizeof(float);   // 174,640 B
  hipFuncSetAttribute(reinterpret_cast<const void*>(nms_kernel),
                      hipFuncAttributeMaxDynamicSharedMemorySize, (int)shmem);
  nms_kernel<<<BATCH, NT2, shmem, stream>>>(x1p, y1p, x2p, y2p, scp, clp, out);
}